// QuanvolutionLayer_64458869178433
// MI455X (gfx1250) — compile-verified
//
#include <hip/hip_runtime.h>
#include <math.h>

// Quanvolution layer, analytically reduced (Heisenberg picture):
//   theta0 = a0 + w0, theta1 = a1 + w1
//   out0 = cos(w2) * cos(theta0)
//   out1 = cos(theta1)
//   out2 = cos(theta0) * cos(a2)
//   out3 = cos(w3) * cos(theta0) * cos(a2) * cos(a3)
// Patch (j,k): a0=x[2j][2k], a1=x[2j][2k+1], a2=x[2j+1][2k], a3=x[2j+1][2k+1]
// Memory-bound: 32MB in + 32MB out @ 23.3 TB/s ~= 2.75 us floor.

#ifndef __has_builtin
#define __has_builtin(x) 0
#endif

#if __has_builtin(__builtin_amdgcn_global_load_async_to_lds_b128)
#define QH_ASYNC_LDS 1
#else
#define QH_ASYNC_LDS 0
#endif

#define IMG_W 512
#define IMG_H 512
#define HP 256
#define WP 256

typedef int v4i __attribute__((ext_vector_type(4)));

__device__ __forceinline__ float cospoly(float x) {
  // cos, Taylor to x^10: |err| < 7e-9 on [-1.1, 1.1]
  float u = x * x;
  float c = fmaf(u, -2.75573192e-7f, 2.48015873e-5f);
  c = fmaf(u, c, -1.38888889e-3f);
  c = fmaf(u, c, 4.16666667e-2f);
  c = fmaf(u, c, -0.5f);
  c = fmaf(u, c, 1.0f);
  return c;
}

__device__ __forceinline__ float sinpoly(float x) {
  // sin (odd), Taylor to x^9: |err| < 8e-8 on [-1.1, 1.1]
  float u = x * x;
  float s = fmaf(u, 2.75573192e-6f, -1.98412698e-4f);
  s = fmaf(u, s, 8.33333333e-3f);
  s = fmaf(u, s, -1.66666667e-1f);
  s = fmaf(u, s, 1.0f);
  return x * s;
}

// Full-range sincos for |w| < ~100 (weights live in [0, 2pi)):
// 2-term Cody-Waite pi/2 reduction + quadrant fixup. ~1e-7 abs error.
__device__ __forceinline__ void sincos_fullrange(float w, float* s, float* c) {
  float k = rintf(w * 0.636619772f);          // 2/pi
  float r = fmaf(-k, 1.57079637e+0f, w);      // pi/2 hi
  r = fmaf(-k, -4.37113883e-8f, r);           // pi/2 lo correction
  float ss = sinpoly(r);
  float cc = cospoly(r);
  int ik = (int)k & 3;
  float sv = (ik & 1) ? cc : ss;
  float cv = (ik & 1) ? -ss : cc;
  if (ik & 2) { sv = -sv; cv = -cv; }
  *s = sv;
  *c = cv;
}

__device__ __forceinline__ void wait_async0() {
#if __has_builtin(__builtin_amdgcn_s_wait_asynccnt)
  __builtin_amdgcn_s_wait_asynccnt(0);
#else
  asm volatile("s_wait_asynccnt 0" ::: "memory");
#endif
}

// One thread = 4 patches (8 input pixels per row x 2 rows). Input is staged
// global -> LDS with CDNA5 async b128 copies (ASYNCcnt), outputs are 4x b128
// stores, one per output channel row segment.
template <bool PRECOMP>
__global__ __launch_bounds__(256) void quanv_main(const float* __restrict__ x,
                                                  const float* __restrict__ wsrc,
                                                  float* __restrict__ out,
                                                  int ngroups) {
  const int g = blockIdx.x * 256 + threadIdx.x;
  if (g >= ngroups) return;

  float cw0, sw0, cw1, sw1, cw2, cw3;
  if (PRECOMP) {
    cw0 = wsrc[0]; sw0 = wsrc[1];
    cw1 = wsrc[2]; sw1 = wsrc[3];
    cw2 = wsrc[4]; cw3 = wsrc[5];
  } else {
    float su;
    sincos_fullrange(wsrc[0], &sw0, &cw0);
    sincos_fullrange(wsrc[1], &sw1, &cw1);
    sincos_fullrange(wsrc[2], &su, &cw2);
    sincos_fullrange(wsrc[3], &su, &cw3);
  }

  const int gx = g & 63;          // group of 4 patches along the row
  const int j  = (g >> 6) & 255;  // patch row
  const int b  = g >> 14;         // batch

  const float* row0 = x + (size_t)b * (IMG_H * IMG_W) + (size_t)(2 * j) * IMG_W + gx * 8;
  const float* row1 = row0 + IMG_W;

  float4 r0a, r0b, r1a, r1b;
#if QH_ASYNC_LDS
  // CDNA5 async-copy path: global -> LDS (bypasses VGPR staging), ASYNCcnt.
  __shared__ float4 stage[256 * 4];
  float4* my = &stage[threadIdx.x * 4];
  __builtin_amdgcn_global_load_async_to_lds_b128((v4i*)(row0),     (v4i*)(my + 0), 0, 0);
  __builtin_amdgcn_global_load_async_to_lds_b128((v4i*)(row0 + 4), (v4i*)(my + 1), 0, 0);
  __builtin_amdgcn_global_load_async_to_lds_b128((v4i*)(row1),     (v4i*)(my + 2), 0, 0);
  __builtin_amdgcn_global_load_async_to_lds_b128((v4i*)(row1 + 4), (v4i*)(my + 3), 0, 0);
  wait_async0();
  asm volatile("" ::: "memory");
  r0a = my[0]; r0b = my[1]; r1a = my[2]; r1b = my[3];
#else
  r0a = *(const float4*)(row0);
  r0b = *(const float4*)(row0 + 4);
  r1a = *(const float4*)(row1);
  r1b = *(const float4*)(row1 + 4);
#endif

  const float pa0[4] = {r0a.x, r0a.z, r0b.x, r0b.z};
  const float pa1[4] = {r0a.y, r0a.w, r0b.y, r0b.w};
  const float pa2[4] = {r1a.x, r1a.z, r1b.x, r1b.z};
  const float pa3[4] = {r1a.y, r1a.w, r1b.y, r1b.w};

  float4 q0, q1, q2, q3;
  float* o0 = (float*)&q0;
  float* o1 = (float*)&q1;
  float* o2 = (float*)&q2;
  float* o3 = (float*)&q3;

#pragma unroll
  for (int p = 0; p < 4; ++p) {
    const float a0 = pa0[p], a1 = pa1[p], a2 = pa2[p], a3 = pa3[p];
    // cos(a + w) = cos a cos w - sin a sin w   (a in [0,1): cheap polynomials)
    const float ct0 = fmaf(cospoly(a0), cw0, -sinpoly(a0) * sw0);
    const float ct1 = fmaf(cospoly(a1), cw1, -sinpoly(a1) * sw1);
    const float c2  = cospoly(a2);
    const float c3  = cospoly(a3);
    o0[p] = cw2 * ct0;
    o1[p] = ct1;
    o2[p] = ct0 * c2;
    o3[p] = cw3 * (ct0 * c2) * c3;
  }

  const size_t ob = (size_t)b * (4 * HP * WP) + (size_t)j * WP + (size_t)(gx * 4);
  *(float4*)(out + ob)               = q0;
  *(float4*)(out + ob + 1 * HP * WP) = q1;
  *(float4*)(out + ob + 2 * HP * WP) = q2;
  *(float4*)(out + ob + 3 * HP * WP) = q3;
}

// Tiny pre-pass: trig of the 4 uniform weights into workspace (one wave,
// ~60 instructions with the custom reduction instead of ocml Payne-Hanek).
__global__ void quanv_wtrig(const float* __restrict__ w, float* __restrict__ wt) {
  if (threadIdx.x == 0 && blockIdx.x == 0) {
    float s0, c0, s1, c1, su, c2, c3;
    sincos_fullrange(w[0], &s0, &c0);
    sincos_fullrange(w[1], &s1, &c1);
    sincos_fullrange(w[2], &su, &c2);
    sincos_fullrange(w[3], &su, &c3);
    wt[0] = c0; wt[1] = s0;
    wt[2] = c1; wt[3] = s1;
    wt[4] = c2; wt[5] = c3;
  }
}

extern "C" void kernel_launch(void* const* d_in, const int* in_sizes, int n_in,
                              void* d_out, int out_size, void* d_ws, size_t ws_size,
                              hipStream_t stream) {
  const float* x = (const float*)d_in[0];   // [B,1,512,512] fp32
  const float* w = (const float*)d_in[1];   // [1,4] fp32
  float* out = (float*)d_out;               // [B,4,256,256] fp32

  const int ngroups = in_sizes[0] / 16;     // 16 pixels (4 patches) per thread
  const int blocks = (ngroups + 255) / 256;

  if (d_ws && ws_size >= 6 * sizeof(float)) {
    float* wtrig = (float*)d_ws;
    quanv_wtrig<<<1, 32, 0, stream>>>(w, wtrig);
    quanv_main<true><<<blocks, 256, 0, stream>>>(x, wtrig, out, ngroups);
  } else {
    quanv_main<false><<<blocks, 256, 0, stream>>>(x, w, out, ngroups);
  }
}